// DifferentiableCDF_7438883357033
// MI455X (gfx1250) — compile-verified
//
#include <hip/hip_runtime.h>

// ---------------------------------------------------------------------------
// DifferentiableCDF (soft Gaussian histogram -> pdf -> cdf) for gfx1250.
//
//   weights = exp(-(x - c)^2 / sigma^2),  sigma = 0.01
//   hist[b,c,bin] = sum_pixels weights
//   cdf = cumsum(hist) / (sum(hist) + 1e-6)
//
// Pass 1: per-workgroup partial histograms. Wave w handles 16 bins; weights
// for (4 pixels x 16 bins) are built per iteration as the B matrix of
// V_WMMA_F32_16X16X4_F32 with A = ones, so the pixel-axis reduction runs on
// the matrix pipe while VALU/TRANS does the exp stream. Pixels are staged
// into LDS with double-buffered GLOBAL_LOAD_ASYNC_TO_LDS (ASYNCcnt) when the
// toolchain exposes the builtins, else a cooperative sync copy.
// Pass 2: reduce partials, inclusive scan, normalize.
// ---------------------------------------------------------------------------

typedef float v2f __attribute__((ext_vector_type(2)));
typedef float v8f __attribute__((ext_vector_type(8)));
typedef int   v2i __attribute__((ext_vector_type(2)));

#define BINS        256
#define NCHAN       12          // B*C = 4*3
#define HW          65536       // 256*256 pixels per channel
#define WG_THREADS  512         // 16 waves of 32 -> 16*16 = 256 bins covered
#define CHUNK       1024        // pixels per LDS buffer (4 KB)

// -1/sigma^2 * log2(e) : exp(-d^2*1e4) == exp2(d^2 * KC)
#define KC (-14426.950408889634f)

#if defined(__has_builtin)
#  if __has_builtin(__builtin_amdgcn_global_load_async_to_lds_b64) && \
      __has_builtin(__builtin_amdgcn_s_wait_asynccnt)
#    define HAVE_ASYNC_LDS 1
#  endif
#endif
#ifndef HAVE_ASYNC_LDS
#  define HAVE_ASYNC_LDS 0
#endif

#if HAVE_ASYNC_LDS
typedef __attribute__((address_space(1))) v2i* gptr_v2i;   // global
typedef __attribute__((address_space(3))) v2i* lptr_v2i;   // LDS

// Each of 512 threads moves 8 bytes -> 4 KB chunk, straight into LDS.
__device__ __forceinline__ void async_copy_chunk(const float* gsrc, float* lds_dst)
{
    const int t = threadIdx.x;
    __builtin_amdgcn_global_load_async_to_lds_b64(
        (gptr_v2i)(gsrc + 2 * t),
        (lptr_v2i)(lds_dst + 2 * t),
        0, 0);
}
#endif

__global__ __launch_bounds__(WG_THREADS)
void cdf_hist_partial_kernel(const float* __restrict__ x,
                             const float* __restrict__ centers,
                             float* __restrict__ partial,
                             int P, int pixelsPerWG)
{
    __shared__ float sx[2][CHUNK];

    const int channel = blockIdx.x / P;
    const int part    = blockIdx.x % P;
    const long base   = (long)channel * HW + (long)part * pixelsPerWG;

    const int lane    = threadIdx.x & 31;
    const int wave    = threadIdx.x >> 5;
    const int n       = lane & 15;          // bin column within this wave's 16
    const int kb      = (lane >> 4) * 2;    // which pixel pair of the K=4 group
    const int binbase = wave * 16;

    const float c = centers[binbase + n];

    v8f acc = {0.f, 0.f, 0.f, 0.f, 0.f, 0.f, 0.f, 0.f};
    v2f a;  a.x = 1.0f;  a.y = 1.0f;        // A matrix: 16x4 of ones

#if HAVE_ASYNC_LDS
    int cur = 0;
    async_copy_chunk(x + base, sx[0]);
    for (int chunk = 0; chunk < pixelsPerWG; chunk += CHUNK) {
        const bool has_next = (chunk + CHUNK) < pixelsPerWG;
        if (has_next)
            async_copy_chunk(x + base + chunk + CHUNK, sx[1 - cur]);
        // our own async loads for the current buffer are the older ones
        if (has_next) __builtin_amdgcn_s_wait_asynccnt(1);
        else          __builtin_amdgcn_s_wait_asynccnt(0);
        __syncthreads();                       // everyone's loads complete

        const float* sc = sx[cur];
        #pragma unroll 4
        for (int p = 0; p < CHUNK; p += 4) {
            const float p0 = sc[p + kb];
            const float p1 = sc[p + kb + 1];
            const float d0 = p0 - c;
            const float d1 = p1 - c;
            v2f b;
            b.x = __builtin_amdgcn_exp2f(d0 * d0 * KC);
            b.y = __builtin_amdgcn_exp2f(d1 * d1 * KC);
            acc = __builtin_amdgcn_wmma_f32_16x16x4_f32(
                false, a, false, b, (short)0, acc, false, false);
        }
        __syncthreads();                       // done reading before next issue
        cur ^= 1;
    }
#else
    for (int chunk = 0; chunk < pixelsPerWG; chunk += CHUNK) {
        for (int i = threadIdx.x; i < CHUNK; i += WG_THREADS)
            sx[0][i] = x[base + chunk + i];
        if ((chunk + CHUNK < pixelsPerWG) && threadIdx.x < 32)
            __builtin_prefetch(&x[base + chunk + CHUNK + threadIdx.x * 32], 0, 3);
        __syncthreads();

        #pragma unroll 4
        for (int p = 0; p < CHUNK; p += 4) {
            const float p0 = sx[0][p + kb];
            const float p1 = sx[0][p + kb + 1];
            const float d0 = p0 - c;
            const float d1 = p1 - c;
            v2f b;
            b.x = __builtin_amdgcn_exp2f(d0 * d0 * KC);
            b.y = __builtin_amdgcn_exp2f(d1 * d1 * KC);
            acc = __builtin_amdgcn_wmma_f32_16x16x4_f32(
                false, a, false, b, (short)0, acc, false, false);
        }
        __syncthreads();
    }
#endif

    // D row M=0 lives in acc[0] of lanes 0..15 (N = lane)
    if (lane < 16)
        partial[(long)blockIdx.x * BINS + binbase + lane] = acc[0];
}

__global__ __launch_bounds__(BINS)
void cdf_finalize_kernel(const float* __restrict__ partial,
                         float* __restrict__ out,
                         int P)
{
    __shared__ float buf[2][BINS];
    const int c = blockIdx.x;
    const int t = threadIdx.x;

    float h = 0.f;
    for (int p = 0; p < P; ++p)
        h += partial[((long)c * P + p) * BINS + t];

    buf[0][t] = h;
    __syncthreads();

    // Hillis-Steele inclusive scan over 256 bins
    int src = 0;
    for (int off = 1; off < BINS; off <<= 1) {
        float v = buf[src][t];
        if (t >= off) v += buf[src][t - off];
        buf[1 - src][t] = v;
        __syncthreads();
        src = 1 - src;
    }

    const float total = buf[src][BINS - 1];       // = sum(hist)
    out[(long)c * BINS + t] = buf[src][t] / (total + 1e-6f);
}

extern "C" void kernel_launch(void* const* d_in, const int* in_sizes, int n_in,
                              void* d_out, int out_size, void* d_ws, size_t ws_size,
                              hipStream_t stream)
{
    (void)in_sizes; (void)n_in; (void)out_size;

    const float* x       = (const float*)d_in[0];   // (4,3,256,256) f32
    const float* centers = (const float*)d_in[1];   // (256,) f32
    float*       out     = (float*)d_out;           // (4,3,256) f32
    float*       partial = (float*)d_ws;

    // P partial histograms per channel, sized to the workspace (power of two).
    int P = 64;
    const size_t perP = (size_t)NCHAN * BINS * sizeof(float);
    while (P > 1 && (size_t)P * perP > ws_size) P >>= 1;
    const int pixelsPerWG = HW / P;

    cdf_hist_partial_kernel<<<dim3(NCHAN * P), dim3(WG_THREADS), 0, stream>>>(
        x, centers, partial, P, pixelsPerWG);
    cdf_finalize_kernel<<<dim3(NCHAN), dim3(BINS), 0, stream>>>(
        partial, out, P);
}